// GCN_6605659701280
// MI455X (gfx1250) — compile-verified
//
#include <hip/hip_runtime.h>
#include <hip/hip_bf16.h>
#include <stdint.h>

// CDNA5 / gfx1250, wave32. f32 WMMA keeps exact fp32 GEMM precision; the
// workload is scatter/atomic bound, so precision is free.

typedef float v2f __attribute__((ext_vector_type(2)));
typedef float v8f __attribute__((ext_vector_type(8)));

// ---------------- utility: zero fill (grid-stride) ----------------
__global__ void zero_f32(float* __restrict__ p, int n) {
  int i = blockIdx.x * blockDim.x + threadIdx.x;
  int stride = gridDim.x * blockDim.x;
  for (; i < n; i += stride) p[i] = 0.0f;
}

// ---------------- degree count: deg[dst[e]] += 1 ----------------
__global__ void count_deg(const long long* __restrict__ dst, float* __restrict__ deg, int E) {
  int e = blockIdx.x * blockDim.x + threadIdx.x;
  if (e < E) atomicAdd(&deg[(int)dst[e]], 1.0f);
}

// ---------------- per-node: dinv = rsqrt(deg+1), selfw = 1/(deg+1) ----------------
__global__ void node_prep(const float* __restrict__ deg, float* __restrict__ dinv,
                          float* __restrict__ selfw, int N) {
  int i = blockIdx.x * blockDim.x + threadIdx.x;
  if (i < N) {
    float d = deg[i] + 1.0f;
    float r = rsqrtf(d);
    dinv[i]  = r;
    selfw[i] = r * r;
  }
}

// ---------------- per-edge norm = dinv[src]*dinv[dst] (shared by both layers) ----------------
__global__ void edge_norm(const long long* __restrict__ src, const long long* __restrict__ dst,
                          const float* __restrict__ dinv, float* __restrict__ norm, int E) {
  int e = blockIdx.x * blockDim.x + threadIdx.x;
  if (e < E) norm[e] = dinv[(int)src[e]] * dinv[(int)dst[e]];
}

// ---------------- H[N,C] = X[N,64] @ W[64,C] via V_WMMA_F32_16X16X4_F32 ----------------
// CT = number of 16-wide column tiles (blockDim = 32*CT). Both A (16x64) and the
// whole W (64 x CT*16, zero-padded past C) are staged in LDS so the inner loop is
// pure ds_load + v_wmma: no divergence, no guarded loads, no 64-bit address math.
//
// ISA 7.12.2 layouts (wave32):
//   A 16x4 f32 : row m = lane&15 (both halves); VGPR0 = K = 2*(lane>>4), VGPR1 = K+1
//   B 4x16 f32 : col n = lane&15;               VGPR0 = K = 2*(lane>>4), VGPR1 = K+1
//   C/D 16x16  : VGPR v -> row v + 8*(lane>>4), col = lane&15
template <int CT>
__global__ void gemm_k64_wmma(const float* __restrict__ X, const float* __restrict__ W,
                              float* __restrict__ H, int N, int C) {
  constexpr int CP = CT * 16;                 // padded column count (compile-time)
  __shared__ float atile[16 * 65];            // +1 pad: conflict-free A reads
  __shared__ float wtile[64 * CP];            // whole weight matrix, zero-padded

  const int row0 = blockIdx.x * 16;

  // cooperative, coalesced load of the 16x64 A tile (32-bit addressing)
  for (int i = threadIdx.x; i < 16 * 64; i += 32 * CT) {
    int r = i >> 6, c = i & 63;
    int gr = row0 + r; if (gr >= N) gr = N - 1;   // clamp (N % 16 == 0 in practice)
    atile[r * 65 + c] = X[gr * 64 + c];
  }
  // stage W with zero padding for columns >= C (removes all guards from hot loop)
  for (int i = threadIdx.x; i < 64 * CP; i += 32 * CT) {
    int r = i / CP, c = i - r * CP;
    wtile[i] = (c < C) ? W[r * C + c] : 0.0f;
  }
  __syncthreads();

  const int wave = threadIdx.x >> 5;
  const int lane = threadIdx.x & 31;
  const int half = lane >> 4;        // selects K-pair within each K=4 step
  const int nl   = lane & 15;
  const int n    = wave * 16 + nl;   // output column (may be >= C only in pad tiles)

  v8f acc = {};
  #pragma unroll
  for (int k0 = 0; k0 < 64; k0 += 4) {
    const int ka = k0 + half * 2;
    v2f a, b;
    a.x = atile[nl * 65 + ka];
    a.y = atile[nl * 65 + ka + 1];
    b.x = wtile[ka * CP + n];
    b.y = wtile[(ka + 1) * CP + n];
    // 8 args: (neg_a, A, neg_b, B, c_mod, C, reuse_a, reuse_b); EXEC is full here.
    acc = __builtin_amdgcn_wmma_f32_16x16x4_f32(false, a, false, b, (short)0, acc,
                                                false, false);
  }

  if (n < C) {
    #pragma unroll
    for (int v = 0; v < 8; ++v) {
      int mrow = row0 + half * 8 + v;
      if (mrow < N) H[mrow * C + n] = acc[v];
    }
  }
}

// ---------------- scatter: AGG[dst[e]*C + f] += H[src[e]*C + f] * norm[e] ----------------
// one thread per (edge, feature); features of an edge are contiguous -> coalesced
// gather of H[src] and L2-resident f32 atomics on AGG (dest arrays << 192MB L2).
// C is a template constant so e = t / C is a mul-shift.
template <int C>
__global__ void scatter_feats(const long long* __restrict__ src, const long long* __restrict__ dst,
                              const float* __restrict__ norm, const float* __restrict__ H,
                              float* __restrict__ AGG, int E) {
  int t = blockIdx.x * blockDim.x + threadIdx.x;   // E*C <= 51.2M, fits int
  if (t >= E * C) return;
  int e = t / C;
  int f = t - e * C;
  int s = (int)src[e];
  int d = (int)dst[e];
  atomicAdd(&AGG[d * C + f], H[s * C + f] * norm[e]);
}

// ---------------- layer-1 epilogue: AGG = relu(AGG + H*selfw[node] + b[f]) (C==64) ---------
__global__ void finish1_relu(float* __restrict__ AGG, const float* __restrict__ H,
                             const float* __restrict__ selfw, const float* __restrict__ b,
                             int N) {
  int i = blockIdx.x * blockDim.x + threadIdx.x;   // N*64 <= 6.4M
  if (i < N * 64) {
    int node = i >> 6, f = i & 63;
    float v = AGG[i] + H[i] * selfw[node] + b[f];
    AGG[i] = v > 0.0f ? v : 0.0f;
  }
}

// ---------------- layer-2 epilogue + row log_softmax, one wave per node ----------------
__global__ void finish2_logsoftmax(const float* __restrict__ AGG, const float* __restrict__ H,
                                   const float* __restrict__ selfw, const float* __restrict__ b,
                                   float* __restrict__ out, int N, int C) {
  int wavesPerBlock = blockDim.x >> 5;
  int node = blockIdx.x * wavesPerBlock + (threadIdx.x >> 5);
  int lane = threadIdx.x & 31;
  if (node >= N) return;                 // uniform per wave

  const float NEG = -3.4e38f;
  float sw = selfw[node];
  int c0 = lane, c1 = lane + 32;
  float v0 = NEG, v1 = NEG;
  int base = node * C;
  if (c0 < C) v0 = AGG[base + c0] + H[base + c0] * sw + b[c0];
  if (c1 < C) v1 = AGG[base + c1] + H[base + c1] * sw + b[c1];

  float m = fmaxf(v0, v1);
  #pragma unroll
  for (int o = 16; o > 0; o >>= 1) m = fmaxf(m, __shfl_xor(m, o, 32));

  float s = 0.0f;
  if (c0 < C) s += expf(v0 - m);
  if (c1 < C) s += expf(v1 - m);
  #pragma unroll
  for (int o = 16; o > 0; o >>= 1) s += __shfl_xor(s, o, 32);

  float lse = m + logf(s);
  if (c0 < C) out[base + c0] = v0 - lse;
  if (c1 < C) out[base + c1] = v1 - lse;
}

// ---------------- host side ----------------
extern "C" void kernel_launch(void* const* d_in, const int* in_sizes, int n_in,
                              void* d_out, int out_size, void* d_ws, size_t ws_size,
                              hipStream_t stream) {
  (void)n_in; (void)out_size; (void)ws_size;

  const float*     x     = (const float*)d_in[0];
  const long long* edges = (const long long*)d_in[1];   // int64 [2, E]
  const float*     W1    = (const float*)d_in[2];
  const float*     b1    = (const float*)d_in[3];
  const float*     W2    = (const float*)d_in[4];
  const float*     b2    = (const float*)d_in[5];
  float*           out   = (float*)d_out;

  const int Fin  = 64;
  const int N    = in_sizes[0] / Fin;     // 100000
  const int E    = in_sizes[1] / 2;       // 800000
  const int HID  = in_sizes[3];           // 64
  const int Cout = in_sizes[5];           // 40

  const long long* src = edges;
  const long long* dst = edges + E;

  // workspace carve-out (256B aligned regions)
  char* ws = (char*)d_ws;
  auto alloc = [&](size_t count) -> float* {
    float* p = (float*)ws;
    ws += ((count * sizeof(float) + 255) / 256) * 256;
    return p;
  };
  float* deg   = alloc((size_t)N);
  float* dinv  = alloc((size_t)N);
  float* selfw = alloc((size_t)N);
  float* norm  = alloc((size_t)E);
  float* h1    = alloc((size_t)N * HID);   // reused as h2 after layer 1 (N*Cout <= N*HID)
  float* agg1  = alloc((size_t)N * HID);   // becomes relu'd hidden activations in-place
  float* agg2  = alloc((size_t)N * Cout);

  // zero accumulators / degree
  zero_f32<<<2048, 256, 0, stream>>>(deg,  N);
  zero_f32<<<2048, 256, 0, stream>>>(agg1, N * HID);
  zero_f32<<<2048, 256, 0, stream>>>(agg2, N * Cout);

  // normalization factors
  count_deg<<<(E + 255) / 256, 256, 0, stream>>>(dst, deg, E);
  node_prep<<<(N + 255) / 256, 256, 0, stream>>>(deg, dinv, selfw, N);
  edge_norm<<<(E + 255) / 256, 256, 0, stream>>>(src, dst, dinv, norm, E);

  const int tileBlocks = (N + 15) / 16;

  // ---- layer 1: h1 = x @ W1 (CT=4 -> 128 threads), scatter, self-loop+bias+relu ----
  gemm_k64_wmma<4><<<tileBlocks, 32 * 4, 0, stream>>>(x, W1, h1, N, HID);
  scatter_feats<64><<<(E * 64 + 255) / 256, 256, 0, stream>>>(src, dst, norm, h1, agg1, E);
  finish1_relu<<<((N * HID) + 255) / 256, 256, 0, stream>>>(agg1, h1, selfw, b1, N);

  // ---- layer 2: h2 = h1' @ W2 (CT=3 -> 96 threads), scatter, epilogue+log_softmax ----
  float* h2 = h1;  // reuse
  gemm_k64_wmma<3><<<tileBlocks, 32 * 3, 0, stream>>>(agg1, W2, h2, N, Cout);
  scatter_feats<40><<<(E * 40 + 255) / 256, 256, 0, stream>>>(src, dst, norm, h2, agg2, E);
  finish2_logsoftmax<<<(N + 7) / 8, 256, 0, stream>>>(agg2, h2, selfw, b2, out, N, Cout);
}